// Attention_11493332484768
// MI455X (gfx1250) — compile-verified
//
#include <hip/hip_runtime.h>
#include <hip/hip_bf16.h>
#include <math.h>

// Problem constants (from reference)
#define B      64
#define T      2000
#define D_MEM  512
#define D_Q    1024
#define D_ATT  128
#define C_LOC  32
#define KCONV  31

typedef float v2f __attribute__((ext_vector_type(2)));
typedef float v4f __attribute__((ext_vector_type(4)));
typedef float v8f __attribute__((ext_vector_type(8)));

// ---------------------------------------------------------------------------
// K0: processed_query = ahs[64,1024] @ Wq[1024,128]  via V_WMMA_F32_16X16X4_F32
// grid (4, 8) tiles, 1 wave each
// ---------------------------------------------------------------------------
__global__ __launch_bounds__(32)
void pq_wmma_kernel(const float* __restrict__ A,   // [B, D_Q]
                    const float* __restrict__ Wq,  // [D_Q, D_ATT]
                    float* __restrict__ pq) {      // [B, D_ATT]
    const int mt   = blockIdx.x;      // 0..3  (M tile)
    const int nt   = blockIdx.y;      // 0..7  (N tile)
    const int lane = threadIdx.x;     // 0..31
    const int half = lane >> 4;       // half-wave selects K pair
    const int l16  = lane & 15;

    v8f acc;
#pragma unroll
    for (int i = 0; i < 8; ++i) acc[i] = 0.0f;

    const float* arow = A + (size_t)(mt * 16 + l16) * D_Q;
    for (int k = 0; k < D_Q; k += 4) {
        const int ka = k + half * 2;
        v2f a, b;
        a.x = arow[ka];
        a.y = arow[ka + 1];
        b.x = Wq[(size_t)ka       * D_ATT + nt * 16 + l16];
        b.y = Wq[(size_t)(ka + 1) * D_ATT + nt * 16 + l16];
        acc = __builtin_amdgcn_wmma_f32_16x16x4_f32(
            false, a, false, b, (short)0, acc, false, false);
    }
#pragma unroll
    for (int r = 0; r < 8; ++r) {
        pq[(size_t)(mt * 16 + r + half * 8) * D_ATT + nt * 16 + l16] = acc[r];
    }
}

// ---------------------------------------------------------------------------
// K1: location conv  aw[B,2,T] (*) Wconv[32,2,31], 'same' pad 15 -> loc[B,32,T]
// One block per (b, 256-t tile). aw tile + halo staged in LDS; each thread
// computes all 32 channels for its t. Wconv indices are block-uniform ->
// scalar loads; LDS reads are stride-1 conflict-free; writes coalesced in t.
// ---------------------------------------------------------------------------
#define CONV_TT 256
#define CONV_HALO (CONV_TT + KCONV - 1)   // 286

__global__ __launch_bounds__(CONV_TT)
void loc_conv_kernel(const float* __restrict__ aw,
                     const float* __restrict__ Wc,
                     const float* __restrict__ bc,
                     float* __restrict__ loc) {
    const int b   = blockIdx.x;
    const int t0  = blockIdx.y * CONV_TT;
    const int tid = threadIdx.x;
    __shared__ float s_aw[2][CONV_HALO];

    for (int o = tid; o < 2 * CONV_HALO; o += CONV_TT) {
        const int i  = o / CONV_HALO;
        const int oo = o % CONV_HALO;
        const int tt = t0 + oo - 15;
        s_aw[i][oo] = (tt >= 0 && tt < T) ? aw[(size_t)(b * 2 + i) * T + tt] : 0.0f;
    }
    __syncthreads();

    const int t = t0 + tid;
    if (t >= T) return;

    for (int c = 0; c < C_LOC; ++c) {
        float s = bc[c];
#pragma unroll
        for (int i = 0; i < 2; ++i) {
            const float* wp = Wc + (size_t)(c * 2 + i) * KCONV;
#pragma unroll
            for (int k = 0; k < KCONV; ++k)
                s = fmaf(s_aw[i][tid + k], wp[k], s);
        }
        loc[((size_t)b * C_LOC + c) * T + t] = s;
    }
}

// ---------------------------------------------------------------------------
// K2: per (b, t-tile of 16):  D = loc_tile[16x32] @ Wloc[32x128] via WMMA,
//     then energies = v . tanh(D + pq + pm) fused on accumulator fragments.
// grid (64, 125), 1 wave per tile. pm is read-once (65 MB) -> NT loads.
// ---------------------------------------------------------------------------
__global__ __launch_bounds__(32)
void energy_wmma_kernel(const float* __restrict__ loc,   // [B, 32, T]
                        const float* __restrict__ Wloc,  // [32, 128]
                        const float* __restrict__ pq,    // [B, 128]
                        const float* __restrict__ pm,    // [B, T, 128]
                        const float* __restrict__ v,     // [128]
                        float* __restrict__ energies) {  // [B, T]
    const int b    = blockIdx.x;
    const int t0   = blockIdx.y * 16;
    const int lane = threadIdx.x;
    const int half = lane >> 4;
    const int l16  = lane & 15;

    v8f acc[8];
#pragma unroll
    for (int j = 0; j < 8; ++j)
#pragma unroll
        for (int i = 0; i < 8; ++i) acc[j][i] = 0.0f;

    const float* locb = loc + (size_t)b * C_LOC * T;
    for (int s = 0; s < 8; ++s) {          // K = 32, 4 per step
        const int ka = s * 4 + half * 2;
        v2f a;
        a.x = locb[(size_t)ka       * T + t0 + l16];
        a.y = locb[(size_t)(ka + 1) * T + t0 + l16];
#pragma unroll
        for (int j = 0; j < 8; ++j) {      // 8 N-tiles covering D_ATT=128
            v2f bb;
            bb.x = Wloc[(size_t)ka       * D_ATT + j * 16 + l16];
            bb.y = Wloc[(size_t)(ka + 1) * D_ATT + j * 16 + l16];
            acc[j] = __builtin_amdgcn_wmma_f32_16x16x4_f32(
                false, a, false, bb, (short)0, acc[j], false, false);
        }
    }

    float vfrag[8], pqfrag[8];
#pragma unroll
    for (int j = 0; j < 8; ++j) {
        vfrag[j]  = v[j * 16 + l16];
        pqfrag[j] = pq[(size_t)b * D_ATT + j * 16 + l16];
    }

    float part[8];
#pragma unroll
    for (int r = 0; r < 8; ++r) part[r] = 0.0f;

#pragma unroll
    for (int r = 0; r < 8; ++r) {
        const int trow = t0 + r + half * 8;
        const float* pmrow = pm + ((size_t)b * T + trow) * D_ATT;
#pragma unroll
        for (int j = 0; j < 8; ++j) {
            const float pmv = __builtin_nontemporal_load(pmrow + j * 16 + l16);
            const float x = acc[j][r] + pqfrag[j] + pmv;
            part[r] = fmaf(tanhf(x), vfrag[j], part[r]);
        }
    }

    // reduce over the 16 lanes of each half-wave (bit 4 = half, untouched)
#pragma unroll
    for (int m = 8; m >= 1; m >>= 1)
#pragma unroll
        for (int r = 0; r < 8; ++r)
            part[r] += __shfl_xor(part[r], m, 32);

    if (l16 == 0) {
#pragma unroll
        for (int r = 0; r < 8; ++r)
            energies[(size_t)b * T + t0 + r + half * 8] = part[r];
    }
}

// ---------------------------------------------------------------------------
// K3: softmax over T per batch -> new_weights (written into d_out region)
// ---------------------------------------------------------------------------
__global__ __launch_bounds__(256)
void softmax_kernel(const float* __restrict__ e, float* __restrict__ w) {
    const int b   = blockIdx.x;
    const int tid = threadIdx.x;
    __shared__ float red[256];
    const float* eb = e + (size_t)b * T;

    float m = -INFINITY;
    for (int t = tid; t < T; t += 256) m = fmaxf(m, eb[t]);
    red[tid] = m; __syncthreads();
    for (int s = 128; s > 0; s >>= 1) {
        if (tid < s) red[tid] = fmaxf(red[tid], red[tid + s]);
        __syncthreads();
    }
    m = red[0]; __syncthreads();

    float sum = 0.0f;
    for (int t = tid; t < T; t += 256) sum += expf(eb[t] - m);
    red[tid] = sum; __syncthreads();
    for (int s = 128; s > 0; s >>= 1) {
        if (tid < s) red[tid] += red[tid + s];
        __syncthreads();
    }
    const float inv = 1.0f / red[0];
    for (int t = tid; t < T; t += 256)
        w[(size_t)b * T + t] = expf(eb[t] - m) * inv;
}

// ---------------------------------------------------------------------------
// K4: context partials [B, NCHUNK, 512], chunked over T (deterministic, no fp
//     atomics). This streams the 262 MB `memory` tensor once -> b128 NT loads,
//     prefetch ahead, 20 chunks for wave parallelism.
// ---------------------------------------------------------------------------
#define NCHUNK 20
#define TCHUNK (T / NCHUNK)
#define D4     (D_MEM / 4)

__global__ __launch_bounds__(D4)
void ctx_partial_kernel(const float* __restrict__ w,
                        const float* __restrict__ mem,
                        float* __restrict__ part) {
    const int b  = blockIdx.x;
    const int ch = blockIdx.y;
    const int d4 = threadIdx.x;          // 0..127, 4 floats each
    const int t0 = ch * TCHUNK;

    v4f acc;
    acc.x = acc.y = acc.z = acc.w = 0.0f;

    const v4f*   base = (const v4f*)(mem + ((size_t)b * T + t0) * D_MEM);
    const float* wb   = w + (size_t)b * T + t0;

    for (int t = 0; t < TCHUNK; ++t) {
        const v4f* row = base + (size_t)t * D4;
        if (t + 8 < TCHUNK)
            __builtin_prefetch((const void*)(row + 8 * D4 + d4), 0, 0);
        const v4f m  = __builtin_nontemporal_load(row + d4);
        const float wt = wb[t];
        acc.x = fmaf(wt, m.x, acc.x);
        acc.y = fmaf(wt, m.y, acc.y);
        acc.z = fmaf(wt, m.z, acc.z);
        acc.w = fmaf(wt, m.w, acc.w);
    }
    ((v4f*)(part + ((size_t)b * NCHUNK + ch) * D_MEM))[d4] = acc;
}

__global__ __launch_bounds__(D4)
void ctx_reduce_kernel(const float* __restrict__ part, float* __restrict__ ctx) {
    const int b  = blockIdx.x;
    const int d4 = threadIdx.x;
    v4f s;
    s.x = s.y = s.z = s.w = 0.0f;
#pragma unroll
    for (int c = 0; c < NCHUNK; ++c) {
        const v4f p = ((const v4f*)(part + ((size_t)b * NCHUNK + c) * D_MEM))[d4];
        s.x += p.x; s.y += p.y; s.z += p.z; s.w += p.w;
    }
    ((v4f*)(ctx + (size_t)b * D_MEM))[d4] = s;
}

// ---------------------------------------------------------------------------
// launch
// ---------------------------------------------------------------------------
extern "C" void kernel_launch(void* const* d_in, const int* in_sizes, int n_in,
                              void* d_out, int out_size, void* d_ws, size_t ws_size,
                              hipStream_t stream) {
    const float* ahs  = (const float*)d_in[0];  // [B,1,D_Q]
    const float* mem  = (const float*)d_in[1];  // [B,T,D_MEM]
    const float* pm   = (const float*)d_in[2];  // [B,T,D_ATT]
    const float* aw   = (const float*)d_in[3];  // [B,2,T]
    const float* Wq   = (const float*)d_in[4];  // [D_Q,D_ATT]
    const float* Wc   = (const float*)d_in[5];  // [C_LOC,2,K]
    const float* bc   = (const float*)d_in[6];  // [C_LOC]
    const float* Wloc = (const float*)d_in[7];  // [C_LOC,D_ATT]
    const float* v    = (const float*)d_in[8];  // [D_ATT]

    float* ctx_out = (float*)d_out;               // [B, D_MEM]
    float* w_out   = ctx_out + (size_t)B * D_MEM; // [B, T]

    // workspace layout (floats)
    float* ws       = (float*)d_ws;
    float* pq       = ws;                                   // B*D_ATT      =   8,192
    float* loc      = pq + (size_t)B * D_ATT;               // B*C_LOC*T    = 4,096,000
    float* energies = loc + (size_t)B * C_LOC * T;          // B*T          =   128,000
    float* part     = energies + (size_t)B * T;             // B*NCHUNK*512 =   655,360

    // K0: query projection (WMMA f32)
    pq_wmma_kernel<<<dim3(B / 16, D_ATT / 16), 32, 0, stream>>>(ahs, Wq, pq);

    // K1: location conv (LDS-tiled)
    loc_conv_kernel<<<dim3(B, (T + CONV_TT - 1) / CONV_TT), CONV_TT, 0, stream>>>(
        aw, Wc, bc, loc);

    // K2: loc projection (WMMA f32) + fused tanh energy
    energy_wmma_kernel<<<dim3(B, T / 16), 32, 0, stream>>>(loc, Wloc, pq, pm, v, energies);

    // K3: softmax -> new_weights (second output)
    softmax_kernel<<<B, 256, 0, stream>>>(energies, w_out);

    // K4/K5: context = weights . memory (first output)
    ctx_partial_kernel<<<dim3(B, NCHUNK), D4, 0, stream>>>(w_out, mem, part);
    ctx_reduce_kernel<<<B, D4, 0, stream>>>(part, ctx_out);
}